// RGIN_60120952209623
// MI455X (gfx1250) — compile-verified
//
#include <hip/hip_runtime.h>
#include <hip/hip_bf16.h>

#define NNODES 100000
#define NEDGES 1600000
#define DIM    128

typedef float v2f __attribute__((ext_vector_type(2)));
typedef float v8f __attribute__((ext_vector_type(8)));

// ---------------------------------------------------------------------------
// Repack weight W[R rows(out) x C cols(in)] (PyTorch Linear layout) into
// k-pair interleaved transposed form:
//   Wp[(k>>1) * 2R + r*2 + (k&1)] = W[r*C + k]
// so the WMMA B operand (b.x = W[n][k], b.y = W[n][k+1]) is one coalesced
// float2 load per lane (lanes n consecutive -> 128B contiguous per half-wave).
// ---------------------------------------------------------------------------
__global__ void transpose_pack(float* __restrict__ dst, const float* __restrict__ src,
                               int R, int C) {
    int i = blockIdx.x * 256 + threadIdx.x;
    if (i >= R * C) return;
    int r = i / C;
    int k = i - r * C;
    dst[(size_t)(k >> 1) * (2 * R) + r * 2 + (k & 1)] = src[i];
}

// h1 = h2 = x  (so the edge atomics accumulate directly into h = x + agg)
__global__ void init_h(float* __restrict__ h1, float* __restrict__ h2,
                       const float* __restrict__ x) {
    size_t i = ((size_t)blockIdx.x * 256 + threadIdx.x) * 4;
    float4 v = *(const float4*)(x + i);
    *(float4*)(h1 + i) = v;
    *(float4*)(h2 + i) = v;
}

// One wave (32 lanes) per edge; lane t owns 4 consecutive floats (float4 gather,
// 512B coalesced per wave). Both edge directions handled in one pass.
__global__ void scatter_add(float* __restrict__ h1, float* __restrict__ h2,
                            const float* __restrict__ x, const int* __restrict__ ei) {
    int wave = threadIdx.x >> 5;
    int lane = threadIdx.x & 31;
    long e = (long)blockIdx.x * 8 + wave;
    if (e >= NEDGES) return;
    int src = ei[e];
    int dst = ei[NEDGES + e];
    int c = lane * 4;
    float4 xs = *(const float4*)(x + (size_t)src * DIM + c);
    float4 xd = *(const float4*)(x + (size_t)dst * DIM + c);
    float* p1 = h1 + (size_t)dst * DIM + c;   // agg over src -> dst
    float* p2 = h2 + (size_t)src * DIM + c;   // flipped edges
    unsafeAtomicAdd(p1 + 0, xs.x);
    unsafeAtomicAdd(p1 + 1, xs.y);
    unsafeAtomicAdd(p1 + 2, xs.z);
    unsafeAtomicAdd(p1 + 3, xs.w);
    unsafeAtomicAdd(p2 + 0, xd.x);
    unsafeAtomicAdd(p2 + 1, xd.y);
    unsafeAtomicAdd(p2 + 2, xd.z);
    unsafeAtomicAdd(p2 + 3, xd.w);
}

// ---------------------------------------------------------------------------
// out[M x Nc] = relu?( A[M x K] * W^T + bias )  via V_WMMA_F32_16X16X4_F32.
// Block = 128 threads = 4 waves; one block owns a 16-row panel; wave w owns
// column tiles w, w+4, ... A panel staged in LDS with +4-float row pad so the
// per-lane b64 A reads hit all 64 banks exactly once (stride K+4 -> 4m+k).
// Lane mapping per the 32-bit A/B VGPR layouts:
//   m = lane&15, hi = lane>>4, kk = 2*hi; a = {A[m][k0+kk], A[m][k0+kk+1]}
//   n = lane&15;                          b = {B[k0+kk][n], B[k0+kk+1][n]}
// D layout: row = v + 8*hi, col = lane&15  (v = accumulator VGPR 0..7).
// ---------------------------------------------------------------------------
template <int K, bool RELU>
__global__ __launch_bounds__(128)
void gemm_wmma(const float* __restrict__ A, const float* __restrict__ Wp,
               const float* __restrict__ bias, float* __restrict__ out, int Nc) {
    constexpr int LDSROW = K + 4;
    __shared__ float As[16 * LDSROW];

    const int row0 = blockIdx.x * 16;
    const int tid  = threadIdx.x;

    // Stage contiguous 16xK A panel into padded LDS (float4, fully coalesced).
    const float* Ap = A + (size_t)row0 * K;
    for (int i = tid; i < 16 * (K / 4); i += 128) {
        int r  = i / (K / 4);
        int c4 = (i - r * (K / 4)) * 4;
        float4 v = *(const float4*)(Ap + r * K + c4);
        float* d = &As[r * LDSROW + c4];
        d[0] = v.x; d[1] = v.y; d[2] = v.z; d[3] = v.w;
    }
    __syncthreads();

    const int wave = tid >> 5;
    const int lane = tid & 31;
    const int n    = lane & 15;
    const int hi   = lane >> 4;
    const int kk   = hi * 2;
    const float* aBase = &As[n * LDSROW + kk];   // A row m == n for this lane

    for (int ct = wave; ct < (Nc >> 4); ct += 4) {
        const int col0 = ct << 4;
        v8f acc = {};
        const float* wptr = Wp + (size_t)hi * 2 * Nc + (size_t)(col0 + n) * 2;
#pragma unroll 8
        for (int k0 = 0; k0 < K; k0 += 4) {
            v2f a = *(const v2f*)(aBase + k0);
            v2f b = *(const v2f*)(wptr);
            acc = __builtin_amdgcn_wmma_f32_16x16x4_f32(
                /*neg_a=*/false, a, /*neg_b=*/false, b,
                /*c_mod=*/(short)0, acc, /*reuse_a=*/false, /*reuse_b=*/false);
            wptr += 4 * Nc;   // advance two k-pairs in packed layout
        }
        float bv  = bias[col0 + n];
        float* op = out + (size_t)row0 * Nc + col0 + n;
#pragma unroll
        for (int v = 0; v < 8; ++v) {
            float r = acc[v] + bv;
            if (RELU) r = fmaxf(r, 0.0f);
            op[(size_t)(v + hi * 8) * Nc] = r;
        }
    }
}

// ret = h + relu(LN(t)*g + beta), written into cat[N x 256] at column colOff.
// One wave per row; lane owns cols {lane, lane+32, lane+64, lane+96}.
__global__ void ln_relu_res(const float* __restrict__ t, const float* __restrict__ h,
                            const float* __restrict__ g, const float* __restrict__ beta,
                            float* __restrict__ cat, int colOff) {
    int wave = threadIdx.x >> 5;
    int lane = threadIdx.x & 31;
    int row  = blockIdx.x * 8 + wave;
    if (row >= NNODES) return;
    const float* tr = t + (size_t)row * DIM;

    float v[4];
#pragma unroll
    for (int i = 0; i < 4; ++i) v[i] = tr[lane + 32 * i];
    float s = v[0] + v[1] + v[2] + v[3];
#pragma unroll
    for (int off = 16; off; off >>= 1) s += __shfl_xor(s, off);
    float mu = s * (1.0f / DIM);

    float d[4], q = 0.0f;
#pragma unroll
    for (int i = 0; i < 4; ++i) { d[i] = v[i] - mu; q += d[i] * d[i]; }
#pragma unroll
    for (int off = 16; off; off >>= 1) q += __shfl_xor(q, off);
    float rs = rsqrtf(q * (1.0f / DIM) + 1e-5f);

    const float* hr = h + (size_t)row * DIM;
    float* cr = cat + (size_t)row * (2 * DIM) + colOff;
#pragma unroll
    for (int i = 0; i < 4; ++i) {
        int c = lane + 32 * i;
        float y = fmaxf(d[i] * rs * g[c] + beta[c], 0.0f);
        cr[c] = hr[c] + y;
    }
}

extern "C" void kernel_launch(void* const* d_in, const int* in_sizes, int n_in,
                              void* d_out, int out_size, void* d_ws, size_t ws_size,
                              hipStream_t stream) {
    (void)in_sizes; (void)n_in; (void)out_size; (void)ws_size;
    const float* x   = (const float*)d_in[0];
    const int*   ei  = (const int*)d_in[1];
    const float* W1  = (const float*)d_in[2];
    const float* b1  = (const float*)d_in[3];
    const float* g1  = (const float*)d_in[4];
    const float* bt1 = (const float*)d_in[5];
    const float* W2  = (const float*)d_in[6];
    const float* b2  = (const float*)d_in[7];
    const float* g2  = (const float*)d_in[8];
    const float* bt2 = (const float*)d_in[9];
    const float* Wl1 = (const float*)d_in[10];
    const float* bl1 = (const float*)d_in[11];
    const float* Wl2 = (const float*)d_in[12];
    const float* bl2 = (const float*)d_in[13];

    float* ws   = (float*)d_ws;
    float* h1   = ws;                                  // N*128
    float* h2   = ws + (size_t)NNODES * DIM;           // N*128
    float* m1   = ws;                                  // N*256 (aliases h1+h2; dead by then)
    float* cat  = ws + (size_t)NNODES * 2 * DIM;       // N*256
    float* wt1  = ws + (size_t)NNODES * 4 * DIM;       // 128*128
    float* wt2  = wt1 + DIM * DIM;                     // 128*128
    float* wl1t = wt2 + DIM * DIM;                     // 256*256
    float* wl2t = wl1t + 2 * DIM * 2 * DIM;            // 128*256
    float* tbuf = (float*)d_out;                       // N*128 GEMM temp, then final out

    transpose_pack<<<(DIM * DIM + 255) / 256, 256, 0, stream>>>(wt1, W1, DIM, DIM);
    transpose_pack<<<(DIM * DIM + 255) / 256, 256, 0, stream>>>(wt2, W2, DIM, DIM);
    transpose_pack<<<(4 * DIM * DIM + 255) / 256, 256, 0, stream>>>(wl1t, Wl1, 2 * DIM, 2 * DIM);
    transpose_pack<<<(2 * DIM * DIM + 255) / 256, 256, 0, stream>>>(wl2t, Wl2, DIM, 2 * DIM);

    init_h<<<(NNODES * DIM) / 1024, 256, 0, stream>>>(h1, h2, x);
    scatter_add<<<NEDGES / 8, 256, 0, stream>>>(h1, h2, x, ei);

    gemm_wmma<128, false><<<NNODES / 16, 128, 0, stream>>>(h1, wt1, b1, tbuf, DIM);
    ln_relu_res<<<NNODES / 8, 256, 0, stream>>>(tbuf, h1, g1, bt1, cat, 0);
    gemm_wmma<128, false><<<NNODES / 16, 128, 0, stream>>>(h2, wt2, b2, tbuf, DIM);
    ln_relu_res<<<NNODES / 8, 256, 0, stream>>>(tbuf, h2, g2, bt2, cat, DIM);

    gemm_wmma<256, true><<<NNODES / 16, 128, 0, stream>>>(cat, wl1t, bl1, m1, 2 * DIM);
    gemm_wmma<256, true><<<NNODES / 16, 128, 0, stream>>>(m1, wl2t, bl2, tbuf, DIM);
}